// DilatedMultiheadSelfAttentionWithRelativePositionalEmbedding_57501022158921
// MI455X (gfx1250) — compile-verified
//
#include <hip/hip_runtime.h>

// ---------------------------------------------------------------------------
// DilatedMultiheadSelfAttentionWithRelativePositionalEmbedding for MI455X.
//  Kernel 1: fp32 WMMA projection GEMM (V_WMMA_F32_16X16X4_F32), 32x32 tile
//            per wave32, K=512 loop.
//  Kernel 2: banded dilated attention (W=6 window), bandwidth-bound, fp32.
// ---------------------------------------------------------------------------

typedef float v2f __attribute__((ext_vector_type(2)));
typedef float v8f __attribute__((ext_vector_type(8)));

#define DMODEL   512
#define NHEADS   8
#define HEADDIM  64
#define WINSZ    6
#define TSEQ     4096
#define BATCH    8
#define MROWS    (BATCH * TSEQ)          // 32768 rows for the projections

// ---------------------------------------------------------------------------
// Kernel 1: Y[M,512] = X[M,512] @ W^T + bias     (nn.Linear semantics)
// One wave computes a 32(M) x 32(N) tile with 4 fp32 WMMA accumulators.
// A-frag (16x4 f32): lane<16 holds row m0+lane, K = k0 + 2*(lane>>4) .. +1
// B-frag (4x16 f32): lane   holds col n0+(lane&15), same half-wave K split.
// ---------------------------------------------------------------------------
__global__ __launch_bounds__(256)
void proj_gemm_wmma_f32(const float* __restrict__ X,
                        const float* __restrict__ W,   // (512 out, 512 in)
                        const float* __restrict__ bias,
                        float* __restrict__ Y)
{
    const int lane = threadIdx.x & 31;
    const int wave = threadIdx.x >> 5;
    const int lr   = lane & 15;
    const int half = lane >> 4;

    const int m0 = blockIdx.y * 256 + wave * 32;   // 8 waves -> 256 rows/block
    const int n0 = blockIdx.x * 32;

    const int rowA0 = m0 + lr;          // M subtile 0
    const int rowA1 = m0 + 16 + lr;     // M subtile 1
    const int colB0 = n0 + lr;          // N subtile 0
    const int colB1 = n0 + 16 + lr;     // N subtile 1

    v8f acc00 = {}; v8f acc01 = {}; v8f acc10 = {}; v8f acc11 = {};

    const float* xa0 = X + (size_t)rowA0 * DMODEL + half * 2;
    const float* xa1 = X + (size_t)rowA1 * DMODEL + half * 2;
    const float* wb0 = W + (size_t)colB0 * DMODEL + half * 2;
    const float* wb1 = W + (size_t)colB1 * DMODEL + half * 2;

#pragma unroll 4
    for (int k0 = 0; k0 < DMODEL; k0 += 4) {
        v2f a0 = *(const v2f*)(xa0 + k0);
        v2f a1 = *(const v2f*)(xa1 + k0);
        v2f b0 = *(const v2f*)(wb0 + k0);
        v2f b1 = *(const v2f*)(wb1 + k0);
        acc00 = __builtin_amdgcn_wmma_f32_16x16x4_f32(false, a0, false, b0,
                                                      (short)0, acc00, false, false);
        acc01 = __builtin_amdgcn_wmma_f32_16x16x4_f32(false, a0, false, b1,
                                                      (short)0, acc01, false, false);
        acc10 = __builtin_amdgcn_wmma_f32_16x16x4_f32(false, a1, false, b0,
                                                      (short)0, acc10, false, false);
        acc11 = __builtin_amdgcn_wmma_f32_16x16x4_f32(false, a1, false, b1,
                                                      (short)0, acc11, false, false);
    }

    const float bb0 = bias[colB0];
    const float bb1 = bias[colB1];

    // D layout: VGPR r holds M = r (lanes 0-15) / M = 8+r (lanes 16-31),
    // N = n_base + (lane & 15).
#pragma unroll
    for (int r = 0; r < 8; ++r) {
        const int row0 = m0 + r + 8 * half;
        const int row1 = m0 + 16 + r + 8 * half;
        Y[(size_t)row0 * DMODEL + n0 + lr]      = acc00[r] + bb0;
        Y[(size_t)row0 * DMODEL + n0 + 16 + lr] = acc01[r] + bb1;
        Y[(size_t)row1 * DMODEL + n0 + lr]      = acc10[r] + bb0;
        Y[(size_t)row1 * DMODEL + n0 + 16 + lr] = acc11[r] + bb1;
    }
}

// ---------------------------------------------------------------------------
// Kernel 2: dilated band attention, one thread per (b, h, t).
// Faithful to kv_roll: indices taken mod (T + pad); pad region is zeros, so
// wrapped / overflowed taps give qk == 0 exactly -> the 1e-9 mask applies.
// ---------------------------------------------------------------------------
__global__ __launch_bounds__(256)
void dilated_attn_f32(const float* __restrict__ Q,
                      const float* __restrict__ K,
                      const float* __restrict__ V,
                      const float* __restrict__ Er,     // (H, 64, 6)
                      const int*   __restrict__ layerPtr,
                      float* __restrict__ Out,          // (B, T, 512)
                      float* __restrict__ AttnOut)      // (B, H, T, 1, 6)
{
    const int gid = blockIdx.x * blockDim.x + threadIdx.x;
    if (gid >= BATCH * NHEADS * TSEQ) return;
    const int t = gid % TSEQ;
    const int h = (gid / TSEQ) % NHEADS;
    const int b = gid / (TSEQ * NHEADS);

    const int layer = layerPtr[0];
    const int dil   = 1 << layer;
    const int pad   = dil * 4;                 // (LEFT_ATTN - 1) * 2**layer
    const int Tp    = TSEQ + pad;
    // round((1+layer)/11 * LATENCY) -- look-ahead tap
    const int o0    = (int)rintf((1.0f + (float)layer) * (10.0f / 11.0f));

    int  idx[WINSZ];
    bool valid[WINSZ];
#pragma unroll
    for (int w = 0; w < WINSZ; ++w) {
        const int off = (w == 0) ? o0 : -((w - 1) * dil);
        int raw = t + off;
        if (raw < 0)        raw += Tp;
        else if (raw >= Tp) raw -= Tp;
        valid[w] = (raw < TSEQ);
        idx[w]   = valid[w] ? raw : 0;         // safe row for guarded loads
    }

    const size_t headOff = (size_t)h * HEADDIM;
    const float4* q4 = (const float4*)(Q + ((size_t)(b * TSEQ + t)) * DMODEL + headOff);
    const float*  er = Er + (size_t)h * HEADDIM * WINSZ;   // + d*6 + w

    // ---- scores -----------------------------------------------------------
    float score[WINSZ];
    float smax = -3.402823466e+38f;
#pragma unroll
    for (int w = 0; w < WINSZ; ++w) {
        const float4* k4 =
            (const float4*)(K + ((size_t)(b * TSEQ + idx[w])) * DMODEL + headOff);
        float qk = 0.0f, rel = 0.0f;
#pragma unroll
        for (int c = 0; c < HEADDIM / 4; ++c) {
            const float4 qv = q4[c];
            const float4 kv = k4[c];
            qk += qv.x * kv.x + qv.y * kv.y + qv.z * kv.z + qv.w * kv.w;
            const int d = 4 * c;
            rel += qv.x * er[(d + 0) * WINSZ + w] + qv.y * er[(d + 1) * WINSZ + w]
                 + qv.z * er[(d + 2) * WINSZ + w] + qv.w * er[(d + 3) * WINSZ + w];
        }
        if (!valid[w]) qk = 0.0f;              // zero-pad rows -> exact 0
        const float mask = (qk == 0.0f) ? 1e-9f : 0.0f;
        float s = (qk + rel) * 0.125f + mask;  // / sqrt(64)
        score[w] = s;
        smax = fmaxf(smax, s);
    }

    // ---- softmax over the 6-tap window ------------------------------------
    float attn[WINSZ];
    float ssum = 0.0f;
#pragma unroll
    for (int w = 0; w < WINSZ; ++w) {
        const float e = expf(score[w] - smax);
        attn[w] = e;
        ssum += e;
    }
    const float inv = 1.0f / ssum;
    float* aout = AttnOut + (((size_t)(b * NHEADS + h) * TSEQ + t)) * WINSZ;
#pragma unroll
    for (int w = 0; w < WINSZ; ++w) {
        attn[w] *= inv;
        aout[w] = attn[w];
    }

    // ---- weighted sum of v rows (zero contribution for padded taps) -------
    const float4* v4[WINSZ];
    float aEff[WINSZ];
#pragma unroll
    for (int w = 0; w < WINSZ; ++w) {
        v4[w] = (const float4*)(V + ((size_t)(b * TSEQ + idx[w])) * DMODEL + headOff);
        aEff[w] = valid[w] ? attn[w] : 0.0f;
    }
    float* op = Out + ((size_t)(b * TSEQ + t)) * DMODEL + headOff;
#pragma unroll
    for (int c = 0; c < HEADDIM / 4; ++c) {
        float4 acc = {0.0f, 0.0f, 0.0f, 0.0f};
#pragma unroll
        for (int w = 0; w < WINSZ; ++w) {
            const float4 vv = v4[w][c];
            acc.x += aEff[w] * vv.x;
            acc.y += aEff[w] * vv.y;
            acc.z += aEff[w] * vv.z;
            acc.w += aEff[w] * vv.w;
        }
        *(float4*)(op + 4 * c) = acc;
    }
}

// ---------------------------------------------------------------------------
extern "C" void kernel_launch(void* const* d_in, const int* in_sizes, int n_in,
                              void* d_out, int out_size, void* d_ws, size_t ws_size,
                              hipStream_t stream)
{
    const float* query = (const float*)d_in[0];
    const float* key   = (const float*)d_in[1];
    const float* value = (const float*)d_in[2];
    const float* Wq    = (const float*)d_in[3];
    const float* bq    = (const float*)d_in[4];
    const float* Wk    = (const float*)d_in[5];
    const float* bk    = (const float*)d_in[6];
    const float* Wv    = (const float*)d_in[7];
    const float* bv    = (const float*)d_in[8];
    const float* Er    = (const float*)d_in[9];
    const int*   layer = (const int*)d_in[10];

    float* out  = (float*)d_out;                                   // (B,T,512)
    float* attn = out + (size_t)MROWS * DMODEL;                    // (B,H,T,1,6)

    const size_t btd = (size_t)MROWS * DMODEL;
    float* qb = (float*)d_ws;
    float* kb = qb + btd;
    float* vb = kb + btd;

    // ---- projections: fp32 WMMA GEMMs ------------------------------------
    dim3 g1(DMODEL / 32, MROWS / 256, 1);
    dim3 b1(256, 1, 1);
    proj_gemm_wmma_f32<<<g1, b1, 0, stream>>>(query, Wq, bq, qb);
    proj_gemm_wmma_f32<<<g1, b1, 0, stream>>>(key,   Wk, bk, kb);
    proj_gemm_wmma_f32<<<g1, b1, 0, stream>>>(value, Wv, bv, vb);

    // ---- dilated attention ------------------------------------------------
    const int total  = BATCH * NHEADS * TSEQ;
    dim3 g2((total + 255) / 256, 1, 1);
    dilated_attn_f32<<<g2, b1, 0, stream>>>(qb, kb, vb, Er, layer, out, attn);
}